// GCN_85761906966870
// MI455X (gfx1250) — compile-verified
//
#include <hip/hip_runtime.h>
#include <math.h>

typedef float v2f __attribute__((ext_vector_type(2)));
typedef float v8f __attribute__((ext_vector_type(8)));

#define WAVE 32
#define TK 32     // K-slice staged in LDS per iteration
#define NT 128    // N-tile width per block/wave for the main GEMM

// ---------------- degree kernels ----------------
__global__ void deg_init(float* __restrict__ deg, int n) {
    int i = blockIdx.x * blockDim.x + threadIdx.x;
    if (i < n) deg[i] = 1.0f;
}

__global__ void deg_edges(const int* __restrict__ dst, const float* __restrict__ w,
                          float* __restrict__ deg, int e) {
    int i = blockIdx.x * blockDim.x + threadIdx.x;
    if (i < e) atomicAdd(&deg[dst[i]], w[i]);
}

__global__ void deg_finish(const float* __restrict__ deg, float* __restrict__ dinv, int n) {
    int i = blockIdx.x * blockDim.x + threadIdx.x;
    if (i < n) dinv[i] = rsqrtf(deg[i]);
}

__global__ void zero_f4(float4* __restrict__ p, long long n4) {
    long long i = (long long)blockIdx.x * blockDim.x + threadIdx.x;
    if (i < n4) p[i] = make_float4(0.f, 0.f, 0.f, 0.f);
}

// ---------------- fp32 WMMA GEMM: C = [relu](A[M,K] @ W[K,N] [+ bias]) ----------------
// block = 256 threads (8 waves); each wave owns a 16 x NT(=128) output strip.
// B staged in LDS with paired-K layout: BsP[p][n] = {W[k0+2p][n0+n], W[k0+2p+1][n0+n]}
// so every B fragment is one aligned ds_load_b64 (no repack movs).
template<bool RELU, bool ADD_BIAS>
__global__ __launch_bounds__(256)
void gemm_wmma_f32(const float* __restrict__ A, const float* __restrict__ W,
                   const float* __restrict__ bias, float* __restrict__ C,
                   int M, int K, int N)
{
    __shared__ float2 BsP[TK / 2][NT];
    const int lane = threadIdx.x & (WAVE - 1);
    const int wv   = threadIdx.x / WAVE;
    const int Mtiles = (M + 15) >> 4;
    const int mt   = blockIdx.y * 8 + wv;
    const int n0   = blockIdx.x * NT;
    const int row  = lane & 15;   // M index within A-tile / N index within B-tile
    const int hi   = lane >> 4;   // lane half selects K pair
    const bool act = (mt < Mtiles);
    const int m0   = act ? (mt << 4) : 0;

    int arow = m0 + row; if (arow > M - 1) arow = M - 1;
    const float* Ap = A + (size_t)arow * K;

    v8f acc[8];
    #pragma unroll
    for (int t = 0; t < 8; ++t) acc[t] = (v8f){};

    for (int k0 = 0; k0 < K; k0 += TK) {
        // cooperatively stage B[k0..k0+TK-1][n0..n0+NT-1] in paired-K form
        for (int i = threadIdx.x; i < (TK / 2) * NT; i += 256) {
            const int p  = i >> 7;        // NT == 128
            const int nn = i & (NT - 1);
            const size_t g = (size_t)(k0 + 2 * p) * N + (n0 + nn);
            BsP[p][nn] = make_float2(W[g], W[g + N]);
        }
        __syncthreads();
        if (act) {
            #pragma unroll
            for (int kk = 0; kk < TK; kk += 4) {
                // A 16x4 f32 layout: reg0 = K=kk (lanes 0-15) / K=kk+2 (lanes 16-31), reg1 = +1
                v2f a = *(const v2f*)(Ap + k0 + kk + 2 * hi);
                const int p = (kk >> 1) + hi;     // K-pair index for this lane half
                const float2* Bp = &BsP[p][row];
                #pragma unroll
                for (int t = 0; t < 8; ++t) {
                    v2f b = *(const v2f*)(&Bp[t * 16]);
                    acc[t] = __builtin_amdgcn_wmma_f32_16x16x4_f32(
                        false, a, false, b, (short)0, acc[t], false, false);
                }
            }
        }
        __syncthreads();
    }
    if (!act) return;

    #pragma unroll
    for (int t = 0; t < 8; ++t) {
        const int col = n0 + t * 16 + row;
        float bb = 0.0f;
        if (ADD_BIAS) bb = bias[col];
        #pragma unroll
        for (int r = 0; r < 8; ++r) {
            const int orow = m0 + r + 8 * hi;  // C/D layout: vgpr r -> M=r (lanes 0-15), M=r+8 (16-31)
            if (orow < M) {
                float v = acc[t][r] + bb;
                if (RELU) v = fmaxf(v, 0.0f);
                C[(size_t)orow * N + col] = v;
            }
        }
    }
}

// ---------------- fused final GEMM (N=64) + bias + log_softmax ----------------
__global__ __launch_bounds__(256)
void gemm_lsm(const float* __restrict__ A, const float* __restrict__ W,
              const float* __restrict__ bias, float* __restrict__ out,
              int M, int K)
{
    __shared__ float2 BsP[TK / 2][64];
    const int lane = threadIdx.x & (WAVE - 1);
    const int wv   = threadIdx.x / WAVE;
    const int Mtiles = (M + 15) >> 4;
    const int mt   = blockIdx.y * 8 + wv;
    const int row  = lane & 15;
    const int hi   = lane >> 4;
    const bool act = (mt < Mtiles);
    const int m0   = act ? (mt << 4) : 0;

    int arow = m0 + row; if (arow > M - 1) arow = M - 1;
    const float* Ap = A + (size_t)arow * K;

    v8f acc[4];
    #pragma unroll
    for (int t = 0; t < 4; ++t) acc[t] = (v8f){};

    for (int k0 = 0; k0 < K; k0 += TK) {
        for (int i = threadIdx.x; i < (TK / 2) * 64; i += 256) {
            const int p  = i >> 6;
            const int nn = i & 63;
            const size_t g = (size_t)(k0 + 2 * p) * 64 + nn;
            BsP[p][nn] = make_float2(W[g], W[g + 64]);
        }
        __syncthreads();
        if (act) {
            #pragma unroll
            for (int kk = 0; kk < TK; kk += 4) {
                v2f a = *(const v2f*)(Ap + k0 + kk + 2 * hi);
                const int p = (kk >> 1) + hi;
                const float2* Bp = &BsP[p][row];
                #pragma unroll
                for (int t = 0; t < 4; ++t) {
                    v2f b = *(const v2f*)(&Bp[t * 16]);
                    acc[t] = __builtin_amdgcn_wmma_f32_16x16x4_f32(
                        false, a, false, b, (short)0, acc[t], false, false);
                }
            }
        }
        __syncthreads();
    }
    if (!act) return;

    float bc[4];
    #pragma unroll
    for (int t = 0; t < 4; ++t) bc[t] = bias[t * 16 + row];

    #pragma unroll
    for (int r = 0; r < 8; ++r) {
        // each lane-half holds one row's 64 logits spread over 16 lanes x 4 accumulators
        float x0 = acc[0][r] + bc[0], x1 = acc[1][r] + bc[1];
        float x2 = acc[2][r] + bc[2], x3 = acc[3][r] + bc[3];
        float mx = fmaxf(fmaxf(x0, x1), fmaxf(x2, x3));
        #pragma unroll
        for (int msk = 8; msk >= 1; msk >>= 1)
            mx = fmaxf(mx, __shfl_xor(mx, msk, WAVE));   // stays within each 16-lane half
        float se = expf(x0 - mx) + expf(x1 - mx) + expf(x2 - mx) + expf(x3 - mx);
        #pragma unroll
        for (int msk = 8; msk >= 1; msk >>= 1)
            se += __shfl_xor(se, msk, WAVE);
        float lse = mx + logf(se);
        int orow = m0 + r + 8 * hi;
        if (orow < M) {
            float* o = out + (size_t)orow * 64;
            o[row +  0] = x0 - lse;
            o[row + 16] = x1 - lse;
            o[row + 32] = x2 - lse;
            o[row + 48] = x3 - lse;
        }
    }
}

// ---------------- edge scatter: agg[dst] += dinv[src]*w*dinv[dst] * L[src] ----------------
__global__ __launch_bounds__(256)
void scatter_edges(const float* __restrict__ L, const int* __restrict__ src,
                   const int* __restrict__ dst, const float* __restrict__ w,
                   const float* __restrict__ dinv, float* __restrict__ agg,
                   int E, int F)
{
    const int e = blockIdx.x * (blockDim.x / WAVE) + (threadIdx.x / WAVE);
    if (e >= E) return;
    const int lane = threadIdx.x & (WAVE - 1);
    const int s = src[e], d = dst[e];
    const float nrm = dinv[s] * w[e] * dinv[d];
    const float4* Ls = (const float4*)(L + (size_t)s * F);
    float* Ad = agg + (size_t)d * F;
    for (int i = lane; i < (F >> 2); i += WAVE) {
        float4 v = Ls[i];
        atomicAdd(&Ad[4 * i + 0], nrm * v.x);
        atomicAdd(&Ad[4 * i + 1], nrm * v.y);
        atomicAdd(&Ad[4 * i + 2], nrm * v.z);
        atomicAdd(&Ad[4 * i + 3], nrm * v.w);
    }
}

// ---------------- combine: H = relu(agg + L/deg + bias), in place on agg ----------------
__global__ __launch_bounds__(256)
void combine_relu(float* __restrict__ agg, const float* __restrict__ L,
                  const float* __restrict__ dinv, const float* __restrict__ bias,
                  int Nn, int F)
{
    long long idx = (long long)blockIdx.x * blockDim.x + threadIdx.x;  // over Nn * F/4
    long long tot = (long long)Nn * (F >> 2);
    if (idx >= tot) return;
    int rown = (int)(idx / (F >> 2));
    int f4   = (int)(idx % (F >> 2));
    float di = dinv[rown];
    float inv = di * di;   // 1/deg
    float4 a = ((const float4*)agg)[idx];
    float4 l = ((const float4*)L)[idx];
    float4 bb = ((const float4*)bias)[f4];
    float4 r;
    r.x = fmaxf(a.x + l.x * inv + bb.x, 0.f);
    r.y = fmaxf(a.y + l.y * inv + bb.y, 0.f);
    r.z = fmaxf(a.z + l.z * inv + bb.z, 0.f);
    r.w = fmaxf(a.w + l.w * inv + bb.w, 0.f);
    ((float4*)agg)[idx] = r;
}

extern "C" void kernel_launch(void* const* d_in, const int* in_sizes, int n_in,
                              void* d_out, int out_size, void* d_ws, size_t ws_size,
                              hipStream_t stream)
{
    (void)n_in; (void)out_size; (void)ws_size;
    const float* x   = (const float*)d_in[0];
    const int*   ei  = (const int*)d_in[1];
    const float* ew  = (const float*)d_in[2];
    const float* W1  = (const float*)d_in[3];
    const float* b1  = (const float*)d_in[4];
    const float* Wc0 = (const float*)d_in[5];
    const float* bc0 = (const float*)d_in[6];
    const float* Wc1 = (const float*)d_in[7];
    const float* bc1 = (const float*)d_in[8];
    const float* W2  = (const float*)d_in[9];
    const float* b2  = (const float*)d_in[10];

    const int F = 256, HD = 256;
    const int Nn = in_sizes[0] / F;      // 100000
    const int E  = in_sizes[2];          // 800000
    const int* srcI = ei;
    const int* dstI = ei + E;

    float* bufA = (float*)d_ws;                       // N*HD
    float* bufB = bufA + (size_t)Nn * HD;             // N*HD
    float* deg  = bufB + (size_t)Nn * HD;             // N
    float* dinv = deg + Nn;                           // N

    const int Mtiles = (Nn + 15) / 16;
    dim3 blk(256);
    dim3 gGemm(HD / NT, (Mtiles + 7) / 8);
    dim3 gLsm(1, (Mtiles + 7) / 8);
    const long long n4 = (long long)Nn * HD / 4;
    const int zBlocks = (int)((n4 + 255) / 256);
    const int eBlocks = (E + 7) / 8;     // 8 edge-waves per block

    // degrees + dinv
    deg_init  <<<(Nn + 255) / 256, blk, 0, stream>>>(deg, Nn);
    deg_edges <<<(E + 255) / 256,  blk, 0, stream>>>(dstI, ew, deg, E);
    deg_finish<<<(Nn + 255) / 256, blk, 0, stream>>>(deg, dinv, Nn);

    // H1 = relu(x @ W1 + b1) -> bufA
    gemm_wmma_f32<true, true><<<gGemm, blk, 0, stream>>>(x, W1, b1, bufA, Nn, F, HD);

    // ---- conv layer 0 ----
    gemm_wmma_f32<false, false><<<gGemm, blk, 0, stream>>>(bufA, Wc0, nullptr, bufB, Nn, HD, HD); // L0
    zero_f4<<<zBlocks, blk, 0, stream>>>((float4*)bufA, n4);
    scatter_edges<<<eBlocks, blk, 0, stream>>>(bufB, srcI, dstI, ew, dinv, bufA, E, HD);
    combine_relu<<<zBlocks, blk, 0, stream>>>(bufA, bufB, dinv, bc0, Nn, HD);                      // H2

    // ---- conv layer 1 ----
    gemm_wmma_f32<false, false><<<gGemm, blk, 0, stream>>>(bufA, Wc1, nullptr, bufB, Nn, HD, HD); // L1
    zero_f4<<<zBlocks, blk, 0, stream>>>((float4*)bufA, n4);
    scatter_edges<<<eBlocks, blk, 0, stream>>>(bufB, srcI, dstI, ew, dinv, bufA, E, HD);
    combine_relu<<<zBlocks, blk, 0, stream>>>(bufA, bufB, dinv, bc1, Nn, HD);                      // H3

    // logits + log_softmax -> d_out
    gemm_lsm<<<gLsm, blk, 0, stream>>>(bufA, W2, b2, (float*)d_out, Nn, HD);
}